// SageMeanAggregator_16758962389080
// MI455X (gfx1250) — compile-verified
//
#include <hip/hip_runtime.h>

// SageMeanAggregator for MI455X (gfx1250), f32 end-to-end.
//   agg = dif_mat[8192x32768] @ feats[dstsrc2src]  -- HBM-bound: streams 1.07 GB once
//   out = relu([agg | feats[dstsrc2dst]] @ w[256x128])
// Matrix math via v_wmma_f32_16x16x4_f32 (wave32 WMMA). Double-buffered LDS
// staging with async-to-LDS copies (ASYNCcnt path), L2 prefetch, and
// non-temporal streaming of dif_mat to protect L2 residency of B.

typedef float v2f __attribute__((ext_vector_type(2)));
typedef float v4f __attribute__((ext_vector_type(4)));
typedef float v8f __attribute__((ext_vector_type(8)));
typedef int   v4i __attribute__((ext_vector_type(4)));

#define N_SRC 32768
#define N_DST 8192
#define DIM   128

#if defined(__has_builtin)
#  if __has_builtin(__builtin_amdgcn_global_load_async_to_lds_b128) && \
      __has_builtin(__builtin_amdgcn_s_wait_asynccnt)
#    define USE_ASYNC_COPY 1
#  endif
#endif
#ifndef USE_ASYNC_COPY
#  define USE_ASYNC_COPY 0
#endif

// ---------------------------------------------------------------------------
// Kernel 1: gather src features into a WMMA-B-friendly row-pair-interleaved
// layout:  Bw[(k>>1)*256 + 2*n + (k&1)] = feats[idx[k]][n]
// so a B fragment {B[k][n], B[k+1][n]} is one aligned 8-byte load.
// Each thread handles a (row-pair, feature-pair): 2x v2f loads -> 1x v4f store.
// ---------------------------------------------------------------------------
__global__ __launch_bounds__(256) void gather_src_interleave_kernel(
    const float* __restrict__ feats,
    const int*   __restrict__ idx,
    float*       __restrict__ Bw)
{
    int tid = blockIdx.x * blockDim.x + threadIdx.x;
    int p   = tid >> 6;            // row pair (2p, 2p+1)
    int d   = (tid & 63) << 1;     // feature pair (d, d+1)
    int r0  = idx[2 * p];
    int r1  = idx[2 * p + 1];
    v2f a = *(const v2f*)(feats + (size_t)r0 * DIM + d);
    v2f b = *(const v2f*)(feats + (size_t)r1 * DIM + d);
    v4f o; o.x = a.x; o.y = b.x; o.z = a.y; o.w = b.y;
    *(v4f*)(Bw + (size_t)p * 256 + 2 * d) = o;   // 16B aligned
}

// ---------------------------------------------------------------------------
// Kernel 2: agg[8192x128] = dif[8192x32768] @ B
// Block: 256 threads (8 waves) -> 64 rows x 128 cols; each wave: 16-col
// stripe, 4 row-tiles (4 WMMAs per B fragment => B L2 traffic ~2.1 GB).
// dif staged through double-buffered LDS (stride 68 -> banks (4r+k)%64 are
// conflict-free b64 A-fragment loads). dif loads are non-temporal; prefetch
// runs 2-3 chunks ahead into L2.
// ---------------------------------------------------------------------------
__global__ __launch_bounds__(256) void agg_gemm_kernel(
    const float* __restrict__ dif,    // [N_DST, N_SRC]
    const float* __restrict__ Bw,     // interleaved [N_SRC/2, 256]
    float*       __restrict__ Cagg)   // [N_DST, DIM]
{
    constexpr int KC     = 64;          // K chunk per LDS stage
    constexpr int LDA    = KC + 4;      // 68 floats padded row stride
    constexpr int BUF    = 64 * LDA;    // one buffer (floats)
    constexpr int NCHUNK = N_SRC / KC;  // 512
    __shared__ float As[2 * BUF];       // 34.8 KB double buffer

    const int tid     = threadIdx.x;
    const int wave    = tid >> 5;
    const int lane    = tid & 31;
    const int ln      = lane & 15;
    const int hi      = lane >> 4;
    const int rowBase = blockIdx.x * 64;
    const int c0      = wave << 4;

    // B fragment pointer (interleaved layout): one b64 per WMMA step
    const float* bptr = Bw + (size_t)hi * 256 + 2 * (c0 + ln);

    v8f acc0 = {}, acc1 = {}, acc2 = {}, acc3 = {};

    // L2 prefetch, 2 and 3 chunks ahead: chunk = 64 rows x 256B = 128 lines;
    // threads 0..127 cover c+2, threads 128..255 cover c+3.
    auto prefetch_ahead = [&](int k0) {
        int t     = tid & 127;
        int ahead = (tid < 128) ? 2 * KC : 3 * KC;
        int kpre  = k0 + ahead;
        if (kpre < N_SRC)
            __builtin_prefetch(dif + (size_t)(rowBase + (t >> 1)) * N_SRC + kpre + (t & 1) * 32, 0, 2);
    };

    auto compute_chunk = [&](const float* buf, int k0) {
        #pragma unroll
        for (int ks = 0; ks < KC; ks += 4) {
            v2f b  = *(const v2f*)(bptr + (((size_t)(k0 + ks)) >> 1) * 256);
            v2f a0 = *(const v2f*)(&buf[(ln)      * LDA + ks + 2 * hi]);
            v2f a1 = *(const v2f*)(&buf[(16 + ln) * LDA + ks + 2 * hi]);
            v2f a2 = *(const v2f*)(&buf[(32 + ln) * LDA + ks + 2 * hi]);
            v2f a3 = *(const v2f*)(&buf[(48 + ln) * LDA + ks + 2 * hi]);
            acc0 = __builtin_amdgcn_wmma_f32_16x16x4_f32(false, a0, false, b, (short)0, acc0, false, false);
            acc1 = __builtin_amdgcn_wmma_f32_16x16x4_f32(false, a1, false, b, (short)0, acc1, false, false);
            acc2 = __builtin_amdgcn_wmma_f32_16x16x4_f32(false, a2, false, b, (short)0, acc2, false, false);
            acc3 = __builtin_amdgcn_wmma_f32_16x16x4_f32(false, a3, false, b, (short)0, acc3, false, false);
        }
    };

#if USE_ASYNC_COPY
    // CDNA5 async copy path: global -> LDS without VGPR round trip (ASYNCcnt).
    // Builtin signature (from hipcc diagnostic): param0 is generic int4*.
    auto stage_async = [&](float* buf, int k0) {
        #pragma unroll
        for (int j = 0; j < 4; ++j) {
            int e = tid + j * 256, r = e >> 4, c4 = (e & 15) << 2;
            const float* g = dif + (size_t)(rowBase + r) * N_SRC + k0 + c4;
            __builtin_amdgcn_global_load_async_to_lds_b128(
                (v4i*)g,
                (__attribute__((address_space(3))) v4i*)&buf[r * LDA + c4],
                0, 0);
        }
    };
    stage_async(As, 0);
    for (int c = 0; c < NCHUNK; ++c) {
        const float* buf = As + (c & 1) * BUF;
        __builtin_amdgcn_s_wait_asynccnt(0);   // chunk c resident in LDS
        __syncthreads();                        // all waves see it; c-1 reads done
        if (c + 1 < NCHUNK)
            stage_async(As + ((c + 1) & 1) * BUF, (c + 1) * KC);
        prefetch_ahead(c * KC);
        compute_chunk(buf, c * KC);
    }
#else
    // Fallback: prefetch chunk into registers (non-temporal), ds_store next iter.
    v4f rbuf[4];
    auto load_chunk = [&](int k0) {
        #pragma unroll
        for (int j = 0; j < 4; ++j) {
            int e = tid + j * 256, r = e >> 4, c4 = (e & 15) << 2;
            rbuf[j] = __builtin_nontemporal_load(
                (const v4f*)(dif + (size_t)(rowBase + r) * N_SRC + k0 + c4));
        }
    };
    auto store_chunk = [&](float* buf) {
        #pragma unroll
        for (int j = 0; j < 4; ++j) {
            int e = tid + j * 256, r = e >> 4, c4 = (e & 15) << 2;
            *(v4f*)(&buf[r * LDA + c4]) = rbuf[j];
        }
    };
    load_chunk(0);
    for (int c = 0; c < NCHUNK; ++c) {
        float* buf = As + (c & 1) * BUF;
        store_chunk(buf);                       // waits on chunk-c loads only
        __syncthreads();
        if (c + 1 < NCHUNK) load_chunk((c + 1) * KC);  // in flight during WMMAs
        prefetch_ahead(c * KC);
        compute_chunk(buf, c * KC);
    }
#endif

    // C/D layout: VGPR v -> row (v + 8*hi), col ln
    #pragma unroll
    for (int v = 0; v < 8; ++v) {
        size_t r = (size_t)rowBase + v + 8 * hi;
        Cagg[(r)      * DIM + c0 + ln] = acc0[v];
        Cagg[(r + 16) * DIM + c0 + ln] = acc1[v];
        Cagg[(r + 32) * DIM + c0 + ln] = acc2[v];
        Cagg[(r + 48) * DIM + c0 + ln] = acc3[v];
    }
}

// ---------------------------------------------------------------------------
// Kernel 3: out = relu( agg @ w[0:128,:] + feats[idx2dst] @ w[128:256,:] )
// 256 threads (8 waves) -> 16 rows x 128 cols; fuses the dst gather.
// ---------------------------------------------------------------------------
__global__ __launch_bounds__(256) void linear_relu_kernel(
    const float* __restrict__ Agg,      // [N_DST, DIM]
    const float* __restrict__ feats,    // [n_nodes, DIM]
    const int*   __restrict__ idx2dst,  // [N_DST]
    const float* __restrict__ W,        // [2*DIM, DIM]
    float*       __restrict__ Out)      // [N_DST, DIM]
{
    const int tid  = threadIdx.x;
    const int wave = tid >> 5;
    const int lane = tid & 31;
    const int ln   = lane & 15;
    const int hi   = lane >> 4;
    const int r0   = blockIdx.x * 16;
    const int c0   = wave << 4;

    const int row     = r0 + ln;
    const float* arow = Agg + (size_t)row * DIM + 2 * hi;
    const int nid     = idx2dst[row];                 // fused dst gather
    const float* drow = feats + (size_t)nid * DIM + 2 * hi;
    const float* wcol = W + (size_t)(2 * hi) * DIM + c0 + ln;

    v8f acc = {};
    #pragma unroll
    for (int k = 0; k < DIM; k += 4) {                // concat half 1: agg
        v2f a = *(const v2f*)(arow + k);
        v2f b;
        b.x = wcol[(size_t)k * DIM];
        b.y = wcol[(size_t)k * DIM + DIM];
        acc = __builtin_amdgcn_wmma_f32_16x16x4_f32(false, a, false, b, (short)0, acc, false, false);
    }
    const float* wcol2 = wcol + (size_t)DIM * DIM;    // w rows 128..255
    #pragma unroll
    for (int k = 0; k < DIM; k += 4) {                // concat half 2: dst feats
        v2f a = *(const v2f*)(drow + k);
        v2f b;
        b.x = wcol2[(size_t)k * DIM];
        b.y = wcol2[(size_t)k * DIM + DIM];
        acc = __builtin_amdgcn_wmma_f32_16x16x4_f32(false, a, false, b, (short)0, acc, false, false);
    }

    #pragma unroll
    for (int v = 0; v < 8; ++v) {
        float x = acc[v];
        Out[(size_t)(r0 + v + 8 * hi) * DIM + c0 + ln] = x > 0.f ? x : 0.f;
    }
}

// ---------------------------------------------------------------------------
extern "C" void kernel_launch(void* const* d_in, const int* in_sizes, int n_in,
                              void* d_out, int out_size, void* d_ws, size_t ws_size,
                              hipStream_t stream)
{
    const float* feats   = (const float*)d_in[0];  // [65536,128] f32
    const int*   idx2src = (const int*)  d_in[1];  // [32768]
    const int*   idx2dst = (const int*)  d_in[2];  // [8192]
    const float* dif     = (const float*)d_in[3];  // [8192,32768] f32
    const float* w       = (const float*)d_in[4];  // [256,128] f32
    float*       out     = (float*)d_out;          // [8192,128] f32

    // workspace: Bw interleaved (16 MB) | agg (4 MB)
    float* Bw  = (float*)d_ws;
    float* agg = (float*)((char*)d_ws + (size_t)N_SRC * DIM * sizeof(float));

    gather_src_interleave_kernel<<<(N_SRC / 2) * (DIM / 2) / 256, 256, 0, stream>>>(feats, idx2src, Bw);
    agg_gemm_kernel             <<<N_DST / 64,                   256, 0, stream>>>(dif, Bw, agg);
    linear_relu_kernel          <<<N_DST / 16,                   256, 0, stream>>>(agg, feats, idx2dst, w, out);
}